// AttnSpanRepr_69750268887128
// MI455X (gfx1250) — compile-verified
//
#include <hip/hip_runtime.h>
#include <math.h>

typedef __attribute__((ext_vector_type(2))) float v2f;
typedef __attribute__((ext_vector_type(8))) float v8f;

#define SEQ 2048
#define BSZ 8
#define HD  1024
#define PD  256
#define NQ  8192
#define M_TOTAL (BSZ*SEQ)   /* 16384 rows */
#define MB  128             /* rows per block */
#define KC  32              /* K chunk staged in LDS */
#define A_STRIDE 36         /* padded floats per A row (mult of 4, bank-friendly) */
#define B_STRIDE 272        /* padded float2 per B k2-row: 544 floats ≡ 32 mod 64 banks */

// ---------------------------------------------------------------------------
// Kernel 1: enc = X @ W + b   (fp32 WMMA 16x16x4), fused logit = enc . attn_w
// ---------------------------------------------------------------------------
__launch_bounds__(256, 1)
__global__ void proj_gemm_kernel(const float* __restrict__ X,    // [16384,1024]
                                 const float* __restrict__ W,    // [1024,256]
                                 const float* __restrict__ pb,   // [256]
                                 const float* __restrict__ aw,   // [256]
                                 const float* __restrict__ ab,   // [1]
                                 float* __restrict__ enc,        // [16384,256]
                                 float* __restrict__ logit)      // [16384]
{
    __shared__ float As[MB * A_STRIDE];
    __shared__ v2f   Bs[(KC/2) * B_STRIDE];
    __shared__ float awS[PD];
    __shared__ float pbS[PD];

    const int t    = threadIdx.x;        // 0..255
    const int wave = t >> 5;             // 0..7
    const int lane = t & 31;
    const int lc   = lane & 15;          // lane%16
    const int half = lane >> 4;          // 0/1
    const int rowBase = blockIdx.x * MB;

    awS[t] = aw[t];
    pbS[t] = pb[t];

    v8f acc[16];
#pragma unroll
    for (int j = 0; j < 16; ++j)
#pragma unroll
        for (int v = 0; v < 8; ++v) acc[j][v] = 0.0f;

    const int arow = wave * 16 + lc;

    for (int k0 = 0; k0 < HD; k0 += KC) {
        // ---- stage A chunk: 128 x 32, coalesced float4 ----
#pragma unroll
        for (int i = 0; i < 4; ++i) {
            int idx = t + 256 * i;                 // 0..1023
            int r   = idx >> 3;
            int c4  = (idx & 7) << 2;
            float4 v = *(const float4*)(X + (size_t)(rowBase + r) * HD + k0 + c4);
            *(float4*)&As[r * A_STRIDE + c4] = v;
        }
        // ---- stage B chunk: K-pair interleaved so B frags are one v2f read ----
#pragma unroll
        for (int i = 0; i < KC/2; ++i) {
            float b0 = W[(size_t)(k0 + 2*i    ) * PD + t];
            float b1 = W[(size_t)(k0 + 2*i + 1) * PD + t];
            v2f p; p.x = b0; p.y = b1;
            Bs[i * B_STRIDE + t] = p;
        }
        // prefetch next chunk while we compute this one (global_prefetch_b8)
        if (k0 + KC < HD) {
            __builtin_prefetch(X + (size_t)(rowBase + (t >> 1)) * HD + k0 + KC, 0, 1);
            __builtin_prefetch(W + (size_t)(k0 + KC + (t >> 4)) * PD + (t & 15) * 16, 0, 1);
        }
        __syncthreads();

        // Hoist all 8 A fragments of this chunk into distinct registers so the
        // inner WMMA bursts never wait on A.
        v2f av[KC/4];
#pragma unroll
        for (int kk = 0; kk < KC/4; ++kk)
            av[kk] = *(const v2f*)&As[arow * A_STRIDE + kk*4 + 2*half];

#pragma unroll
        for (int kk = 0; kk < KC/4; ++kk) {
            // Preload the 16 B fragments of this k-step into distinct temps:
            // lets the scheduler issue all ds_loads back-to-back and cover
            // them with one dscnt wait instead of wait-per-pair.
            v2f bf[16];
#pragma unroll
            for (int j = 0; j < 16; ++j)
                bf[j] = Bs[(kk*2 + half) * B_STRIDE + j*16 + lc];
#pragma unroll
            for (int j = 0; j < 16; ++j)
                acc[j] = __builtin_amdgcn_wmma_f32_16x16x4_f32(
                    false, av[kk], false, bf[j], (short)0, acc[j], false, false);
        }
        __syncthreads();
    }

    // ---- epilogue: bias add, store enc, fused logit dot + 16-lane reduction ----
    const int row0 = rowBase + wave * 16;
    float pv[8];
#pragma unroll
    for (int v = 0; v < 8; ++v) pv[v] = 0.0f;

#pragma unroll
    for (int j = 0; j < 16; ++j) {
        int col = j * 16 + lc;
        float bias = pbS[col];
        float wcol = awS[col];
#pragma unroll
        for (int v = 0; v < 8; ++v) {
            // C/D layout: VGPR v, lanes0-15 -> M=v, lanes16-31 -> M=v+8, N=lane%16
            float e = acc[j][v] + bias;
            enc[(size_t)(row0 + v + 8*half) * PD + col] = e;
            pv[v] += e * wcol;
        }
    }
    float abv = ab[0];
#pragma unroll
    for (int v = 0; v < 8; ++v) {
        float p = pv[v];
#pragma unroll
        for (int o = 8; o > 0; o >>= 1) p += __shfl_xor(p, o, 16);
        if (lc == 0) logit[row0 + v + 8*half] = p + abv;
    }
}

// ---------------------------------------------------------------------------
// Kernel 2: one wave32 per (query, span-set); lane-parallel masked softmax,
// then weighted sum of <=32 enc rows (256 ch, 8 per lane, float4 loads).
// ---------------------------------------------------------------------------
__launch_bounds__(256, 1)
__global__ void span_attn_kernel(const float* __restrict__ enc,    // [16384,256]
                                 const float* __restrict__ logit,  // [16384]
                                 const int* __restrict__ s1, const int* __restrict__ e1,
                                 const int* __restrict__ s2, const int* __restrict__ e2,
                                 const int* __restrict__ qb,
                                 float* __restrict__ out)          // [2*8192,256]
{
    const int wid  = (blockIdx.x * blockDim.x + threadIdx.x) >> 5;  // 0..16383
    const int lane = threadIdx.x & 31;
    const int set  = wid >> 13;
    const int q    = wid & (NQ - 1);

    const int s = set ? s2[q] : s1[q];
    const int e = set ? e2[q] : e1[q];
    const int b = qb[q];

    const int  tok   = s + lane;
    const bool valid = (tok <= e);
    float lg = valid ? logit[b * SEQ + tok] : -3.0e38f;

    float m = lg;
#pragma unroll
    for (int o = 16; o > 0; o >>= 1) m = fmaxf(m, __shfl_xor(m, o));
    float w = valid ? expf(lg - m) : 0.0f;   // masked lanes: w = 0 directly, no exp taken
    float sum = w;
#pragma unroll
    for (int o = 16; o > 0; o >>= 1) sum += __shfl_xor(sum, o);
    float wn = w / sum;

    float a0 = 0.f, a1 = 0.f, a2 = 0.f, a3 = 0.f, a4 = 0.f, a5 = 0.f, a6 = 0.f, a7 = 0.f;
    const float* base = enc + ((size_t)b * SEQ + s) * PD + lane * 8;
    const int len = e - s + 1;
    for (int tt = 0; tt < len; ++tt) {
        float wt = __shfl(wn, tt);
        float4 x0 = *(const float4*)(base + (size_t)tt * PD);
        float4 x1 = *(const float4*)(base + (size_t)tt * PD + 4);
        a0 = fmaf(wt, x0.x, a0); a1 = fmaf(wt, x0.y, a1);
        a2 = fmaf(wt, x0.z, a2); a3 = fmaf(wt, x0.w, a3);
        a4 = fmaf(wt, x1.x, a4); a5 = fmaf(wt, x1.y, a5);
        a6 = fmaf(wt, x1.z, a6); a7 = fmaf(wt, x1.w, a7);
    }
    float* o = out + (size_t)wid * PD + lane * 8;   // wid == set*8192 + q
    *(float4*)(o)     = make_float4(a0, a1, a2, a3);
    *(float4*)(o + 4) = make_float4(a4, a5, a6, a7);
}

// ---------------------------------------------------------------------------
extern "C" void kernel_launch(void* const* d_in, const int* in_sizes, int n_in,
                              void* d_out, int out_size, void* d_ws, size_t ws_size,
                              hipStream_t stream) {
    (void)in_sizes; (void)n_in; (void)out_size; (void)ws_size;
    // setup_inputs order:
    // 0 flag, 1 encoded_input, 2 start_ids_1, 3 end_ids_1, 4 query_batch_idx,
    // 5 start_ids_2, 6 end_ids_2, 7 proj_w, 8 proj_b, 9 attn_w, 10 attn_b
    const float* X  = (const float*)d_in[1];
    const int*   s1 = (const int*)  d_in[2];
    const int*   e1 = (const int*)  d_in[3];
    const int*   qb = (const int*)  d_in[4];
    const int*   s2 = (const int*)  d_in[5];
    const int*   e2 = (const int*)  d_in[6];
    const float* W  = (const float*)d_in[7];
    const float* pb = (const float*)d_in[8];
    const float* aw = (const float*)d_in[9];
    const float* ab = (const float*)d_in[10];

    float* enc   = (float*)d_ws;                       // 16384*256 floats (16 MB)
    float* logit = enc + (size_t)M_TOTAL * PD;         // 16384 floats
    float* out   = (float*)d_out;                      // res1 || res2

    hipLaunchKernelGGL(proj_gemm_kernel, dim3(M_TOTAL / MB), dim3(256), 0, stream,
                       X, W, pb, aw, ab, enc, logit);
    hipLaunchKernelGGL(span_attn_kernel, dim3((2 * NQ) / 8), dim3(256), 0, stream,
                       enc, logit, s1, e1, s2, e2, qb, out);
}